// PolicyHead_68685116998378
// MI455X (gfx1250) — compile-verified
//
#include <hip/hip_runtime.h>

typedef __attribute__((ext_vector_type(2))) float v2f;
typedef __attribute__((ext_vector_type(4))) float v4f;
typedef __attribute__((ext_vector_type(8))) float v8f;

#define B_DIM 1024
#define H_DIM 128
#define E_DIM 16
#define V_DIM 100000L
#define K_DIM 512

// ---------------------------------------------------------------------------
// Kernel 1: query = h @ W^T + b   via V_WMMA_F32_16X16X4_F32
// One wave (32 threads) computes a 16x16 tile: 16 batch rows x 16 (=E) cols,
// accumulating over K=H=128 in 32 steps of K=4.
// ---------------------------------------------------------------------------
__global__ __launch_bounds__(32) void query_proj_wmma(
    const float* __restrict__ h,     // [B,H]
    const float* __restrict__ W,     // [E,H]
    const float* __restrict__ bias,  // [E]
    float* __restrict__ query)       // [B,E]
{
    const int lane  = threadIdx.x;          // 0..31, full wave, EXEC all-1s
    const int r0    = blockIdx.x * 16;      // batch-row tile base
    const int mn    = lane & 15;            // A: row M / B: col N
    const int khalf = (lane >> 4) * 2;      // lanes 0-15 -> K{0,1}; 16-31 -> K{2,3}

    const float* hrow = h + (long)(r0 + mn) * H_DIM;  // A-matrix row (M = mn)
    const float* wrow = W + (long)mn * H_DIM;         // B-matrix: B[k][n] = W[n][k]

    v8f c = {};
    #pragma unroll 4
    for (int k = 0; k < H_DIM; k += 4) {
        v2f a, bm;
        a.x  = hrow[k + khalf];
        a.y  = hrow[k + khalf + 1];
        bm.x = wrow[k + khalf];
        bm.y = wrow[k + khalf + 1];
        // (neg_a, A, neg_b, B, c_mod, C, reuse_a, reuse_b)
        c = __builtin_amdgcn_wmma_f32_16x16x4_f32(false, a, false, bm,
                                                  (short)0, c, false, false);
    }

    const float bn    = bias[mn];            // N = mn
    const int   mbase = (lane >> 4) * 8;     // C/D: M = vgpr + (lane>=16 ? 8:0)
    #pragma unroll
    for (int j = 0; j < 8; ++j) {
        query[(long)(r0 + mbase + j) * E_DIM + mn] = c[j] + bn;
    }
}

// ---------------------------------------------------------------------------
// Kernel 2: fill d_out with -inf. 409.6 MB of pure stores = the roofline cost
// (~17.6 us at 23.3 TB/s). Non-temporal b128 stores: output >> 192 MB L2,
// keep L2 for the embedding table used by the scatter pass.
// ---------------------------------------------------------------------------
__global__ __launch_bounds__(256) void fill_neg_inf(
    v4f* __restrict__ out, long n4)
{
    long i      = (long)blockIdx.x * blockDim.x + threadIdx.x;
    long stride = (long)gridDim.x * blockDim.x;
    const float ninf = -__builtin_inff();
    const v4f v = {ninf, ninf, ninf, ninf};
    for (; i < n4; i += stride) {
        __builtin_nontemporal_store(v, out + i);
    }
}

// ---------------------------------------------------------------------------
// Kernel 3: scatter the 512 valid scores per row.
// out[b, idx] = dot(query[b,:], emb[idx,:])   (16-term dot, 64B gathers,
// emb table is 6.4 MB -> L2-resident; query row uniform per block -> L0 hit)
// Duplicate idx -> identical value -> benign race.
// ---------------------------------------------------------------------------
__global__ __launch_bounds__(256) void scatter_scores(
    const float* __restrict__ query,        // [B,E]
    const int*   __restrict__ valid_indices,// [B,K]
    const float* __restrict__ emb,          // [V,E]
    float* __restrict__ out)                // [B,V]
{
    const int b = blockIdx.x >> 1;                       // 2 blocks per row
    const int k = ((blockIdx.x & 1) << 8) + threadIdx.x; // 0..511

    const v4f* q4 = (const v4f*)(query + (long)b * E_DIM);
    const v4f q0 = q4[0], q1 = q4[1], q2 = q4[2], q3 = q4[3];

    const int idx = valid_indices[(long)b * K_DIM + k];

    const v4f* e4 = (const v4f*)(emb + (long)idx * E_DIM);
    const v4f e0 = e4[0], e1 = e4[1], e2 = e4[2], e3 = e4[3];

    float d = 0.0f;
    d = fmaf(q0.x, e0.x, d); d = fmaf(q0.y, e0.y, d);
    d = fmaf(q0.z, e0.z, d); d = fmaf(q0.w, e0.w, d);
    d = fmaf(q1.x, e1.x, d); d = fmaf(q1.y, e1.y, d);
    d = fmaf(q1.z, e1.z, d); d = fmaf(q1.w, e1.w, d);
    d = fmaf(q2.x, e2.x, d); d = fmaf(q2.y, e2.y, d);
    d = fmaf(q2.z, e2.z, d); d = fmaf(q2.w, e2.w, d);
    d = fmaf(q3.x, e3.x, d); d = fmaf(q3.y, e3.y, d);
    d = fmaf(q3.z, e3.z, d); d = fmaf(q3.w, e3.w, d);

    out[(long)b * V_DIM + idx] = d;
}

extern "C" void kernel_launch(void* const* d_in, const int* in_sizes, int n_in,
                              void* d_out, int out_size, void* d_ws, size_t ws_size,
                              hipStream_t stream) {
    const float* h    = (const float*)d_in[0];   // [1024,128]
    const int*   vidx = (const int*)  d_in[1];   // [1024,512]
    const float* emb  = (const float*)d_in[2];   // [100000,16]
    const float* W    = (const float*)d_in[3];   // [16,128]
    const float* bias = (const float*)d_in[4];   // [16]
    float* out   = (float*)d_out;                // [1024,100000]
    float* query = (float*)d_ws;                 // [1024,16] scratch (64 KB)

    // 1) WMMA query projection: 64 tiles of 16 rows, one wave each.
    query_proj_wmma<<<B_DIM / 16, 32, 0, stream>>>(h, W, bias, query);

    // 2) Fill all of d_out with -inf (bandwidth-bound floor: ~17.6 us).
    const long n4 = (long)B_DIM * V_DIM / 4;     // 25.6M float4
    fill_neg_inf<<<4096, 256, 0, stream>>>((v4f*)out, n4);

    // 3) Scatter the valid scores (stream-ordered after the fill).
    scatter_scores<<<B_DIM * 2, 256, 0, stream>>>(query, vidx, emb, out);
}